// SAGEConvEncoder_29807073034302
// MI455X (gfx1250) — compile-verified
//
#include <hip/hip_runtime.h>

#define N_NODES 100000
#define D 128
#define LDS_STRIDE 132   // 128 + 4 pad: avoids 64-bank conflicts, keeps float4 alignment
#define TILE_M 80        // nodes per block (100000 = 80 * 1250, exact)
#define NSUB 5           // TILE_M / 16 M-subtiles per wave

typedef __attribute__((ext_vector_type(2))) float v2f;
typedef __attribute__((ext_vector_type(8))) float v8f;

// ---------------------------------------------------------------- zero fill
__global__ void zero_f4(float4* __restrict__ p, long n4) {
  long i = (long)blockIdx.x * blockDim.x + threadIdx.x;
  if (i < n4) p[i] = make_float4(0.f, 0.f, 0.f, 0.f);
}

// ------------------------------------------- weight transpose (done once)
// WT[n*128 + k] = W[k*128 + n] for all four weight matrices; grid 64 x 256.
__global__ void transpose_w(const float* __restrict__ W0, const float* __restrict__ W1,
                            const float* __restrict__ W2, const float* __restrict__ W3,
                            float* __restrict__ T0, float* __restrict__ T1,
                            float* __restrict__ T2, float* __restrict__ T3) {
  int t = blockIdx.x * 256 + threadIdx.x;   // 0 .. 16383
  int k = t >> 7;
  int n = t & 127;
  int o = n * D + k;
  T0[o] = W0[t];
  T1[o] = W1[t];
  T2[o] = W2[t];
  T3[o] = W3[t];
}

// ------------------------------------------------- edge scatter (mean prep)
// One wave32 per edge: 128 floats -> 4 coalesced global_atomic_add_f32/lane.
__global__ void scatter_accum(const float* __restrict__ feat,
                              const int* __restrict__ src,
                              const int* __restrict__ dst,
                              float* __restrict__ summed,
                              float* __restrict__ cnt,
                              int n_edges, int do_cnt) {
  int wave = threadIdx.x >> 5;
  int lane = threadIdx.x & 31;
  int e = blockIdx.x * 8 + wave;
  if (e >= n_edges) return;
  int s = src[e];
  int d = dst[e];
  const float* fs = feat + (long)s * D;
  float* sd = summed + (long)d * D;
#pragma unroll
  for (int k = 0; k < 4; ++k) {
    int c = lane + 32 * k;
    atomicAdd(&sd[c], fs[c]);
  }
  if (do_cnt && lane == 0) atomicAdd(&cnt[d], 1.0f);
}

// ------------------------------------------------------ fused SAGE linear
// out[i,:] = (summed[i,:]/max(cnt[i],1)) @ Wl + bl + xin[i,:] @ Wr  (opt ReLU)
// 256 threads = 8 waves; 80 nodes/block; wave w owns cols [16w,16w+16) and
// 5 M-subtiles sharing each B fragment. Weights pre-transposed so each lane's
// (K,K+1) pair is one contiguous global_load_b64.
__global__ void __launch_bounds__(256)
sage_linear(const float* __restrict__ summed,
            const float* __restrict__ cnt,
            const float* __restrict__ xin,
            const float* __restrict__ WlT,   // [N=128][K=128] transposed
            const float* __restrict__ bl,
            const float* __restrict__ WrT,   // [N=128][K=128] transposed
            float* __restrict__ out,
            int relu) {
  __shared__ __align__(16) float sAggr[TILE_M * LDS_STRIDE];
  __shared__ __align__(16) float sX[TILE_M * LDS_STRIDE];

  const int  t    = threadIdx.x;
  const long base = (long)blockIdx.x * TILE_M;

  // Stage TILE_Mx128 aggr (scaled by 1/deg) and self features into LDS.
#pragma unroll
  for (int i = 0; i < (TILE_M * 32) / 256; ++i) {   // 10 iters
    int idx = t + i * 256;
    int r   = idx >> 5;           // node row 0..79
    int c4  = idx & 31;           // float4 column
    long node = base + r;
    float inv = 1.0f / fmaxf(cnt[node], 1.0f);
    float4 sv = *(const float4*)(summed + node * D + c4 * 4);
    float4 av;
    av.x = sv.x * inv; av.y = sv.y * inv; av.z = sv.z * inv; av.w = sv.w * inv;
    *(float4*)(sAggr + r * LDS_STRIDE + c4 * 4) = av;
    *(float4*)(sX    + r * LDS_STRIDE + c4 * 4) =
        *(const float4*)(xin + node * D + c4 * 4);
  }
  __syncthreads();

  const int wave  = t >> 5;
  const int lane  = t & 31;
  const int m     = lane & 15;            // A row (M) / C col-local (N)
  const int khalf = (lane >> 4) * 2;      // lanes 16-31 hold K+2,K+3
  const int col   = wave * 16 + m;        // global output column

  float bias = bl[col];
  v8f acc[NSUB];
#pragma unroll
  for (int s = 0; s < NSUB; ++s)
#pragma unroll
    for (int r = 0; r < 8; ++r) acc[s][r] = bias;

  const float* BL = WlT + (long)col * D;  // this lane's column, K-contiguous
  const float* BR = WrT + (long)col * D;

  // aggr @ Wl
#pragma unroll 4
  for (int k = 0; k < D; k += 4) {
    v2f b = *(const v2f*)(BL + k + khalf);
#pragma unroll
    for (int s = 0; s < NSUB; ++s) {
      const float* ap = sAggr + (s * 16 + m) * LDS_STRIDE + k + khalf;
      v2f a; a.x = ap[0]; a.y = ap[1];
      acc[s] = __builtin_amdgcn_wmma_f32_16x16x4_f32(false, a, false, b,
                                                     (short)0, acc[s], false, false);
    }
  }
  // x @ Wr
#pragma unroll 4
  for (int k = 0; k < D; k += 4) {
    v2f b = *(const v2f*)(BR + k + khalf);
#pragma unroll
    for (int s = 0; s < NSUB; ++s) {
      const float* ap = sX + (s * 16 + m) * LDS_STRIDE + k + khalf;
      v2f a; a.x = ap[0]; a.y = ap[1];
      acc[s] = __builtin_amdgcn_wmma_f32_16x16x4_f32(false, a, false, b,
                                                     (short)0, acc[s], false, false);
    }
  }

  // C/D layout: VGPR r -> M=r (lanes 0-15) or M=r+8 (lanes 16-31), N=lane&15.
  const int rowhi = (lane >> 4) * 8;
#pragma unroll
  for (int s = 0; s < NSUB; ++s)
#pragma unroll
    for (int r = 0; r < 8; ++r) {
      long row = base + s * 16 + r + rowhi;
      float v = acc[s][r];
      if (relu) v = fmaxf(v, 0.0f);
      out[row * D + col] = v;
    }
}

// ----------------------------------------------------------------- launcher
extern "C" void kernel_launch(void* const* d_in, const int* in_sizes, int n_in,
                              void* d_out, int out_size, void* d_ws, size_t ws_size,
                              hipStream_t stream) {
  const float* x   = (const float*)d_in[0];
  const int*   ei  = (const int*)d_in[1];
  const float* Wl0 = (const float*)d_in[2];
  const float* bl0 = (const float*)d_in[3];
  const float* Wr0 = (const float*)d_in[4];
  const float* Wl1 = (const float*)d_in[5];
  const float* bl1 = (const float*)d_in[6];
  const float* Wr1 = (const float*)d_in[7];

  const int E = in_sizes[1] / 2;
  const int* src = ei;
  const int* dst = ei + E;

  float* outp  = (float*)d_out;                  // doubles as scatter accumulator
  float* h     = (float*)d_ws;                   // hidden activations [N, 128]
  float* cnt   = h + (long)N_NODES * D;          // degrees [N]
  float* WlT0  = cnt + N_NODES;                  // 4 transposed weight matrices
  float* WrT0  = WlT0 + D * D;
  float* WlT1  = WrT0 + D * D;
  float* WrT1  = WlT1 + D * D;

  const long n4_feat = (long)N_NODES * D / 4;
  const long n4_cnt  = N_NODES / 4;
  const int  sblocks = (E + 7) / 8;
  const int  gblocks = N_NODES / TILE_M;         // 1250, exact

  // One-time weight transposes (cheap; same stream, so it runs before the zero fills).
  transpose_w<<<(D * D) / 256, 256, 0, stream>>>(Wl0, Wr0, Wl1, Wr1,
                                                 WlT0, WrT0, WlT1, WrT1);

  // Layer 0
  zero_f4<<<(n4_feat + 255) / 256, 256, 0, stream>>>((float4*)outp, n4_feat);
  zero_f4<<<(n4_cnt  + 255) / 256, 256, 0, stream>>>((float4*)cnt,  n4_cnt);
  scatter_accum<<<sblocks, 256, 0, stream>>>(x, src, dst, outp, cnt, E, 1);
  sage_linear<<<gblocks, 256, 0, stream>>>(outp, cnt, x, WlT0, bl0, WrT0, h, 1);

  // Layer 1 (degrees reused; accumulate h, final result into d_out in place)
  zero_f4<<<(n4_feat + 255) / 256, 256, 0, stream>>>((float4*)outp, n4_feat);
  scatter_accum<<<sblocks, 256, 0, stream>>>(h, src, dst, outp, cnt, E, 0);
  sage_linear<<<gblocks, 256, 0, stream>>>(outp, cnt, h, WlT1, bl1, WrT1, outp, 0);
}